// RegressionModel_51135880626627
// MI455X (gfx1250) — compile-verified
//
#include <hip/hip_runtime.h>
#include <hip/hip_bf16.h>
#include <math.h>

#define HF 9           // hidden feature count
#define FIN 16         // input feature count

typedef __attribute__((ext_vector_type(2))) float v2f;
typedef __attribute__((ext_vector_type(8))) float v8f;

// ---------------------------------------------------------------------------
// WMMA linear: out[n, OUT_F] = in[n, 0:IN_F] @ W[IN_F, OUT_F] (+ bias)
// One wave (32 lanes) computes one 16-row x 16-col tile using
// V_WMMA_F32_16X16X4_F32 chained over K in steps of 4.
// A frag (16x4 f32): lanes 0-15 hold K={k0,k0+1}, lanes 16-31 K={k0+2,k0+3}
// B frag (4x16 f32): mirrored K striping, N = lane&15
// C/D (16x16 f32):  vgpr r -> M = r + 8*(lane>=16), N = lane&15
// All fragment loads are unconditional with clamped (always in-bounds)
// addresses; invalid elements are zeroed with selects -> no exec branching
// around the WMMA chain. Stores take a wave-uniform full-tile fast path.
// ---------------------------------------------------------------------------
template <int IN_F, int OUT_F, bool HAS_BIAS>
__global__ void wmma_linear_kernel(const float* __restrict__ in,
                                   const float* __restrict__ W,
                                   const float* __restrict__ bias,
                                   float* __restrict__ out,
                                   int nrows) {
  const int lane = threadIdx.x & 31;
  const int wave = threadIdx.x >> 5;
  const int tile = blockIdx.x * (blockDim.x >> 5) + wave;
  const int row0 = tile * 16;
  if (row0 >= nrows) return;  // wave-uniform: EXEC stays all-1s for WMMA

  const int half = lane >> 4;   // 0 = lanes 0-15, 1 = lanes 16-31
  const int lid  = lane & 15;
  constexpr int KPAD  = (IN_F + 3) & ~3;
  constexpr int NFRAG = KPAD / 4;

  // ---- A fragments (clamped row, clamped k, select-zeroed) ----
  const int  m      = row0 + lid;
  const bool mvalid = (m < nrows);
  const size_t rowoff = (size_t)(mvalid ? m : (nrows - 1)) * IN_F;

  v2f afrag[NFRAG];
#pragma unroll
  for (int k0 = 0; k0 < KPAD; k0 += 4) {
    const int kA = k0 + half * 2;
    const float x0 = in[rowoff + ((kA + 0) < IN_F ? (kA + 0) : 0)];
    const float x1 = in[rowoff + ((kA + 1) < IN_F ? (kA + 1) : 0)];
    afrag[k0 / 4].x = (mvalid && (kA + 0) < IN_F) ? x0 : 0.0f;
    afrag[k0 / 4].y = (mvalid && (kA + 1) < IN_F) ? x1 : 0.0f;
  }

  // ---- B fragments (weights are tiny; clamped loads, select-zeroed) ----
  const int  n   = lid;
  const bool nv  = (n < OUT_F);
  const int  ncl = nv ? n : 0;

  v2f bfrag[NFRAG];
#pragma unroll
  for (int k0 = 0; k0 < KPAD; k0 += 4) {
    const int kB = k0 + half * 2;
    const float w0 = W[(size_t)((kB + 0) < IN_F ? (kB + 0) : 0) * OUT_F + ncl];
    const float w1 = W[(size_t)((kB + 1) < IN_F ? (kB + 1) : 0) * OUT_F + ncl];
    bfrag[k0 / 4].x = (nv && (kB + 0) < IN_F) ? w0 : 0.0f;
    bfrag[k0 / 4].y = (nv && (kB + 1) < IN_F) ? w1 : 0.0f;
  }

  // ---- Bias: one clamped load, added unconditionally (stores are guarded) --
  const float bv = HAS_BIAS ? bias[ncl] : 0.0f;

  // ---- K-chained WMMA accumulation ----
  v8f acc = {};
#pragma unroll
  for (int k0 = 0; k0 < KPAD; k0 += 4) {
    acc = __builtin_amdgcn_wmma_f32_16x16x4_f32(
        /*neg_a=*/false, afrag[k0 / 4], /*neg_b=*/false, bfrag[k0 / 4],
        /*c_mod=*/(short)0, acc, /*reuse_a=*/false, /*reuse_b=*/false);
  }

  // ---- Stores ----
  if (row0 + 16 <= nrows) {
    // Full tile (common case): only the loop-invariant column mask diverges.
    if (nv) {
      float* p = out + (size_t)(row0 + half * 8) * OUT_F + n;
#pragma unroll
      for (int r = 0; r < 8; ++r)
        p[(size_t)r * OUT_F] = acc[r] + bv;
    }
  } else {
    // Boundary tile: per-row guard.
#pragma unroll
    for (int r = 0; r < 8; ++r) {
      const int mr = row0 + r + half * 8;
      if (mr < nrows && nv)
        out[(size_t)mr * OUT_F + n] = acc[r] + bv;
    }
  }
}

// ---------------------------------------------------------------------------
// Per-node prep: attention scalars, init segment-max/sum/agg, zero stats.
// ---------------------------------------------------------------------------
__global__ void node_prep_kernel(const float* __restrict__ h,
                                 const float* __restrict__ att_src,
                                 const float* __restrict__ att_dst,
                                 float* __restrict__ a_src,
                                 float* __restrict__ a_dst,
                                 float* __restrict__ mmax,
                                 float* __restrict__ ssum,
                                 float* __restrict__ agg,
                                 float* __restrict__ stats,
                                 int n) {
  const int i = blockIdx.x * blockDim.x + threadIdx.x;
  if (i < 32) stats[i] = 0.0f;
  if (i >= n) return;
  float s = 0.0f, d = 0.0f;
#pragma unroll
  for (int f = 0; f < HF; ++f) {
    const float v = h[(size_t)i * HF + f];
    s += v * att_src[f];
    d += v * att_dst[f];
  }
  a_src[i] = s;
  a_dst[i] = d;
  mmax[i] = -INFINITY;
  ssum[i] = 0.0f;
#pragma unroll
  for (int f = 0; f < HF; ++f) agg[(size_t)i * HF + f] = 0.0f;
}

__device__ __forceinline__ void atomicMaxF(float* addr, float v) {
  // Works for mixed-sign values when initialized to -inf.
  if (v >= 0.0f)
    atomicMax((int*)addr, __float_as_int(v));
  else
    atomicMin((unsigned int*)addr, __float_as_uint(v));
}

__device__ __forceinline__ float leaky02(float x) {
  return x > 0.0f ? x : 0.2f * x;
}

#define PREF_AHEAD 4096  // elements ahead on the streaming index arrays

// Pass 1 over edges (+ N virtual self-loop edges): segment max of attention.
__global__ void edge_max_kernel(const int* __restrict__ ei,
                                const float* __restrict__ a_src,
                                const float* __restrict__ a_dst,
                                float* __restrict__ mmax,
                                int E, int n) {
  const int i = blockIdx.x * blockDim.x + threadIdx.x;
  if (i >= E + n) return;
  int s, d;
  if (i < E) {
    __builtin_prefetch(&ei[i + PREF_AHEAD], 0, 0);
    __builtin_prefetch(&ei[E + i + PREF_AHEAD], 0, 0);
    s = ei[i];
    d = ei[E + i];
  } else {
    s = i - E;
    d = s;
  }
  const float e = leaky02(a_src[s] + a_dst[d]);
  atomicMaxF(&mmax[d], e);
}

// Pass 2: ex = exp(e - m[dst]); accumulate ex into ssum and ex*h[src] into agg.
// (alpha normalization deferred: out[d] = agg[d] / (ssum[d] + eps))
__global__ void edge_agg_kernel(const int* __restrict__ ei,
                                const float* __restrict__ a_src,
                                const float* __restrict__ a_dst,
                                const float* __restrict__ mmax,
                                const float* __restrict__ h,
                                float* __restrict__ ssum,
                                float* __restrict__ agg,
                                int E, int n) {
  const int i = blockIdx.x * blockDim.x + threadIdx.x;
  if (i >= E + n) return;
  int s, d;
  if (i < E) {
    __builtin_prefetch(&ei[i + PREF_AHEAD], 0, 0);
    __builtin_prefetch(&ei[E + i + PREF_AHEAD], 0, 0);
    s = ei[i];
    d = ei[E + i];
  } else {
    s = i - E;
    d = s;
  }
  const float e  = leaky02(a_src[s] + a_dst[d]);
  const float ex = __expf(e - mmax[d]);
  atomicAdd(&ssum[d], ex);
#pragma unroll
  for (int f = 0; f < HF; ++f)
    atomicAdd(&agg[(size_t)d * HF + f], ex * h[(size_t)s * HF + f]);
}

// Normalize by softmax denominator, add bias, ReLU, accumulate BN statistics.
__global__ void finalize_relu_stats_kernel(const float* __restrict__ agg,
                                           const float* __restrict__ ssum,
                                           const float* __restrict__ bias,
                                           float* __restrict__ hout,
                                           float* __restrict__ stats,
                                           int n) {
  __shared__ float sh[32];  // [0..15]=sum, [16..31]=sumsq
  const int tid = threadIdx.x;
  if (tid < 32) sh[tid] = 0.0f;
  __syncthreads();
  const int i = blockIdx.x * blockDim.x + tid;
  if (i < n) {
    const float sc = 1.0f / (ssum[i] + 1e-16f);
#pragma unroll
    for (int f = 0; f < HF; ++f) {
      float v = agg[(size_t)i * HF + f] * sc + bias[f];
      v = v > 0.0f ? v : 0.0f;
      hout[(size_t)i * HF + f] = v;
      atomicAdd(&sh[f], v);
      atomicAdd(&sh[16 + f], v * v);
    }
  }
  __syncthreads();
  if (tid < HF) {
    atomicAdd(&stats[tid], sh[tid]);
    atomicAdd(&stats[16 + tid], sh[16 + tid]);
  }
}

// Apply batch-norm (biased variance, matching x.var(0)).
__global__ void bn_apply_kernel(float* __restrict__ h,
                                const float* __restrict__ stats,
                                const float* __restrict__ gamma,
                                const float* __restrict__ beta,
                                int n) {
  const int i = blockIdx.x * blockDim.x + threadIdx.x;
  if (i >= n) return;
  const float invn = 1.0f / (float)n;
#pragma unroll
  for (int f = 0; f < HF; ++f) {
    const float mu  = stats[f] * invn;
    const float var = stats[16 + f] * invn - mu * mu;
    const float v = (h[(size_t)i * HF + f] - mu) * rsqrtf(var + 1e-5f);
    h[(size_t)i * HF + f] = gamma[f] * v + beta[f];
  }
}

__global__ void zero_kernel(float* __restrict__ p, int n) {
  const int i = blockIdx.x * blockDim.x + threadIdx.x;
  if (i < n) p[i] = 0.0f;
}

__global__ void pool_kernel(const float* __restrict__ h,
                            const int* __restrict__ batch,
                            float* __restrict__ pooled,
                            int n) {
  const int i = blockIdx.x * blockDim.x + threadIdx.x;
  if (i >= n) return;
  const int b = batch[i];
#pragma unroll
  for (int f = 0; f < HF; ++f)
    atomicAdd(&pooled[(size_t)b * HF + f], h[(size_t)i * HF + f]);
}

// ---------------------------------------------------------------------------
extern "C" void kernel_launch(void* const* d_in, const int* in_sizes, int n_in,
                              void* d_out, int out_size, void* d_ws, size_t ws_size,
                              hipStream_t stream) {
  const float* x     = (const float*)d_in[0];
  const int*   ei    = (const int*)d_in[1];
  const int*   batch = (const int*)d_in[2];
  const float* W1  = (const float*)d_in[3];
  const float* as1 = (const float*)d_in[4];
  const float* ad1 = (const float*)d_in[5];
  const float* b1  = (const float*)d_in[6];
  const float* g1  = (const float*)d_in[7];
  const float* be1 = (const float*)d_in[8];
  const float* W2  = (const float*)d_in[9];
  const float* as2 = (const float*)d_in[10];
  const float* ad2 = (const float*)d_in[11];
  const float* b2  = (const float*)d_in[12];
  const float* g2  = (const float*)d_in[13];
  const float* be2 = (const float*)d_in[14];
  const float* mW1 = (const float*)d_in[15];
  const float* mb1 = (const float*)d_in[16];
  const float* mW2 = (const float*)d_in[17];
  const float* mb2 = (const float*)d_in[18];
  const float* mW3 = (const float*)d_in[19];
  const float* mb3 = (const float*)d_in[20];
  const float* mW4 = (const float*)d_in[21];
  const float* mb4 = (const float*)d_in[22];

  const int N = in_sizes[0] / FIN;
  const int E = in_sizes[1] / 2;
  const int G = out_size;  // output is [G, 1]

  // Workspace layout (floats)
  float* ws      = (float*)d_ws;
  float* featX   = ws;                          // [N, HF] layer input/output feats
  float* featH   = featX + (size_t)N * HF;      // [N, HF] post-linear feats
  float* featAgg = featH + (size_t)N * HF;      // [N, HF] aggregation buffer
  float* a_src   = featAgg + (size_t)N * HF;    // [N]
  float* a_dst   = a_src + N;                   // [N]
  float* mmax    = a_dst + N;                   // [N]
  float* ssum    = mmax + N;                    // [N]
  float* stats   = ssum + N;                    // [32]
  float* pooled  = stats + 32;                  // [G, HF]
  float* t1      = pooled + (size_t)G * HF;     // [G, HF]
  float* t2      = t1 + (size_t)G * HF;         // [G, HF]

  const int BLK = 256;
  const int nodeGrid = (N + BLK - 1) / BLK;
  const int edgeGrid = (E + N + BLK - 1) / BLK;
  auto linGrid = [&](int rows) {
    const int tiles = (rows + 15) / 16;
    return (tiles + 7) / 8;  // 8 waves (tiles) per 256-thread block
  };

  // ----- GAT layer 1 -----
  wmma_linear_kernel<FIN, HF, false><<<linGrid(N), BLK, 0, stream>>>(x, W1, nullptr, featH, N);
  node_prep_kernel<<<nodeGrid, BLK, 0, stream>>>(featH, as1, ad1, a_src, a_dst,
                                                 mmax, ssum, featAgg, stats, N);
  edge_max_kernel<<<edgeGrid, BLK, 0, stream>>>(ei, a_src, a_dst, mmax, E, N);
  edge_agg_kernel<<<edgeGrid, BLK, 0, stream>>>(ei, a_src, a_dst, mmax, featH,
                                                ssum, featAgg, E, N);
  finalize_relu_stats_kernel<<<nodeGrid, BLK, 0, stream>>>(featAgg, ssum, b1,
                                                           featX, stats, N);
  bn_apply_kernel<<<nodeGrid, BLK, 0, stream>>>(featX, stats, g1, be1, N);

  // ----- GAT layer 2 -----
  wmma_linear_kernel<HF, HF, false><<<linGrid(N), BLK, 0, stream>>>(featX, W2, nullptr, featH, N);
  node_prep_kernel<<<nodeGrid, BLK, 0, stream>>>(featH, as2, ad2, a_src, a_dst,
                                                 mmax, ssum, featAgg, stats, N);
  edge_max_kernel<<<edgeGrid, BLK, 0, stream>>>(ei, a_src, a_dst, mmax, E, N);
  edge_agg_kernel<<<edgeGrid, BLK, 0, stream>>>(ei, a_src, a_dst, mmax, featH,
                                                ssum, featAgg, E, N);
  finalize_relu_stats_kernel<<<nodeGrid, BLK, 0, stream>>>(featAgg, ssum, b2,
                                                           featX, stats, N);
  bn_apply_kernel<<<nodeGrid, BLK, 0, stream>>>(featX, stats, g2, be2, N);

  // ----- Pool + MLP head -----
  zero_kernel<<<(G * HF + BLK - 1) / BLK, BLK, 0, stream>>>(pooled, G * HF);
  pool_kernel<<<nodeGrid, BLK, 0, stream>>>(featX, batch, pooled, N);

  wmma_linear_kernel<HF, HF, true><<<linGrid(G), BLK, 0, stream>>>(pooled, mW1, mb1, t1, G);
  wmma_linear_kernel<HF, HF, true><<<linGrid(G), BLK, 0, stream>>>(t1, mW2, mb2, t2, G);
  wmma_linear_kernel<HF, HF, true><<<linGrid(G), BLK, 0, stream>>>(t2, mW3, mb3, t1, G);
  wmma_linear_kernel<HF, 1, true><<<linGrid(G), BLK, 0, stream>>>(t1, mW4, mb4, (float*)d_out, G);
}